// CLUBv2_14422500180346
// MI455X (gfx1250) — compile-verified
//
#include <hip/hip_runtime.h>
#include <hip/hip_bf16.h>

typedef __attribute__((ext_vector_type(2))) float v2f;
typedef __attribute__((ext_vector_type(8))) float v8f;

#define CLUB_N 1024
#define CLUB_D 256
#define ROWG   8          // row groups (each covers 128 rows)
#define BETA_F 0.001f

// Kernel 1: per-column partial sums (S) and sums of squares (Q) via fp32 WMMA.
// grid = 16 column-blocks * ROWG row-groups = 128 blocks, 256 threads (8 waves).
// Wave w of block (cb, rb) reduces rows [rb*128 + w*16, +16) of columns
// [cb*16, +16) using V_WMMA_F32_16X16X4_F32 with an all-ones A matrix, so
// D[m,n] = sum_k B[k,n] regardless of the K-slot permutation.
__global__ void club_colstats_wmma(const float* __restrict__ y,
                                   float* __restrict__ wsS,
                                   float* __restrict__ wsQ) {
    const int cb   = blockIdx.x & 15;   // column block: 16 columns
    const int rb   = blockIdx.x >> 4;   // row group: 128 rows
    const int tid  = threadIdx.x;
    const int w    = tid >> 5;          // wave id in block (0..7)
    const int lane = tid & 31;
    const int col  = (cb << 4) + (lane & 15);
    const int rsel = (lane < 16) ? 0 : 2;       // K-slot split across lane halves
    const int rowbase = rb * 128 + w * 16;

    v2f a;                               // A(16x4) all ones in every lane/VGPR
    a.x = 1.0f; a.y = 1.0f;
    v8f accS = {};                       // column sums
    v8f accQ = {};                       // column sums of squares

#pragma unroll
    for (int s = 0; s < 4; ++s) {        // 4 K-steps of 4 rows = 16 rows
        const int r = rowbase + 4 * s + rsel;
        v2f b, b2;
        b.x = y[r * CLUB_D + col];
        b.y = y[(r + 1) * CLUB_D + col];
        b2.x = b.x * b.x;
        b2.y = b.y * b.y;
        // (neg_a, A, neg_b, B, c_mod, C, reuse_a, reuse_b)
        accS = __builtin_amdgcn_wmma_f32_16x16x4_f32(
            false, a, false, b, (short)0, accS, false, false);
        accQ = __builtin_amdgcn_wmma_f32_16x16x4_f32(
            false, a, false, b2, (short)0, accQ, false, false);
    }

    // Every accumulator element equals the column sum (all A rows are ones);
    // acc[0] at lane L holds the sum for column (L mod 16).
    __shared__ float ldsS[8][16];
    __shared__ float ldsQ[8][16];
    if (lane < 16) {
        ldsS[w][lane] = accS[0];
        ldsQ[w][lane] = accQ[0];
    }
    __syncthreads();

    if (tid < 16) {
        float s = 0.0f, q = 0.0f;
#pragma unroll
        for (int i = 0; i < 8; ++i) { s += ldsS[i][tid]; q += ldsQ[i][tid]; }
        wsS[rb * CLUB_D + (cb << 4) + tid] = s;
        wsQ[rb * CLUB_D + (cb << 4) + tid] = q;
    }
}

// Kernel 2: finish. result = BETA * sum_d ( Q_d/N - S_d^2/N^2 ).
// Fixed-order tree reduction -> deterministic across replays.
__global__ void club_finish(const float* __restrict__ wsS,
                            const float* __restrict__ wsQ,
                            float* __restrict__ out) {
    const int t = threadIdx.x;   // column d, 0..255
    float S = 0.0f, Q = 0.0f;
#pragma unroll
    for (int rb = 0; rb < ROWG; ++rb) {
        S += wsS[rb * CLUB_D + t];
        Q += wsQ[rb * CLUB_D + t];
    }
    const float invN = 1.0f / (float)CLUB_N;
    float val = Q * invN - (S * invN) * (S * invN);

    __shared__ float red[256];
    red[t] = val;
    __syncthreads();
    for (int s = 128; s > 0; s >>= 1) {
        if (t < s) red[t] += red[t + s];
        __syncthreads();
    }
    if (t == 0) out[0] = BETA_F * red[0];
}

extern "C" void kernel_launch(void* const* d_in, const int* in_sizes, int n_in,
                              void* d_out, int out_size, void* d_ws, size_t ws_size,
                              hipStream_t stream) {
    (void)in_sizes; (void)n_in; (void)out_size; (void)ws_size;
    const float* y = (const float*)d_in[0];
    float* wsS = (float*)d_ws;                 // ROWG*256 floats
    float* wsQ = wsS + ROWG * CLUB_D;          // ROWG*256 floats (16 KB total)

    club_colstats_wmma<<<16 * ROWG, 256, 0, stream>>>(y, wsS, wsQ);
    club_finish<<<1, 256, 0, stream>>>(wsS, wsQ, (float*)d_out);
}